// GridSample_18588618457299
// MI455X (gfx1250) — compile-verified
//
#include <hip/hip_runtime.h>
#include <stdint.h>
#include <limits.h>

// ---------------------------------------------------------------------------
// Voxel grid sample (CDNA5 / gfx1250, wave32)
//   1) per-cloud int-coord min/max (LDS + global atomics)
//   2) composite 56-bit keys: (voxelKey << 23) | originalIndex  (unique keys
//      => unstable-safe total order identical to stable argsort tie-break)
//   3) 5-pass LSD radix sort, 12-bit digits, per-(digit,wave) LDS counters,
//      wave32 ballot ranking for stable per-pass scatter. Key buffers +
//      histogram (~160MB) are L2-resident on MI455X (192MB L2).
//   4) boundary compaction -> points / coords / num_unique.
// TDM (tensor_load_to_lds) stages each 32KB key tile into LDS in the scatter
// kernel; keys are consumed twice from LDS (count phase + rank phase).
// ---------------------------------------------------------------------------

#define B_CLOUDS 4
#define N_PTS    2000000
#define BN       (B_CLOUDS * N_PTS)              // 8,000,000
#define TILE     4096
#define NUM_TILES ((BN + TILE - 1) / TILE)       // 1954
#define PADDED_N  (NUM_TILES * TILE)             // 8,003,584
#define RADIX_BITS 12
#define RADIX      (1 << RADIX_BITS)             // 4096
#define IDX_BITS   23
#define IDX_MASK   ((1u << IDX_BITS) - 1u)
#define VSIZE      0.05f

#define SC_T 128          // scatter threads (4 waves)
#define SC_W 4            // waves per scatter block
#define SC_I 32           // items per thread in scatter (4*32*32 = 4096)

typedef unsigned int       u32;
typedef unsigned long long u64;

typedef unsigned int u32x4 __attribute__((ext_vector_type(4)));
typedef int          i32x8 __attribute__((ext_vector_type(8)));
typedef int          i32x4 __attribute__((ext_vector_type(4)));

#if defined(__has_builtin)
#if __has_builtin(__builtin_amdgcn_tensor_load_to_lds) && __has_builtin(__builtin_amdgcn_s_wait_tensorcnt)
#define HAVE_TDM 1
#endif
#endif

// ---------------------------------------------------------------- pass 0: minmax
__global__ __launch_bounds__(64) void init_kernel(int* minc, int* maxc) {
    int t = threadIdx.x;
    if (t < B_CLOUDS * 3) { minc[t] = INT_MAX; maxc[t] = INT_MIN; }
}

__global__ __launch_bounds__(256) void minmax_kernel(const float4* __restrict__ pts4,
                                                     int* __restrict__ minc,
                                                     int* __restrict__ maxc) {
    __shared__ int smn[3], smx[3];
    const int tid = threadIdx.x;
    const int b   = blockIdx.y;
    if (tid < 3) { smn[tid] = INT_MAX; smx[tid] = INT_MIN; }
    __syncthreads();

    const float4* p = pts4 + (size_t)b * N_PTS;
    size_t base = (size_t)blockIdx.x * TILE;
    int mn0 = INT_MAX, mn1 = INT_MAX, mn2 = INT_MAX;
    int mx0 = INT_MIN, mx1 = INT_MIN, mx2 = INT_MIN;
#pragma unroll
    for (int it = 0; it < 16; ++it) {
        size_t i = base + (size_t)it * 256 + tid;
        if (i < (size_t)N_PTS) {
            float4 v = p[i];
            int c0 = (int)(v.x / VSIZE);   // trunc-toward-zero, matches .astype(int64)
            int c1 = (int)(v.y / VSIZE);
            int c2 = (int)(v.z / VSIZE);
            mn0 = min(mn0, c0); mx0 = max(mx0, c0);
            mn1 = min(mn1, c1); mx1 = max(mx1, c1);
            mn2 = min(mn2, c2); mx2 = max(mx2, c2);
        }
    }
    atomicMin(&smn[0], mn0); atomicMax(&smx[0], mx0);
    atomicMin(&smn[1], mn1); atomicMax(&smx[1], mx1);
    atomicMin(&smn[2], mn2); atomicMax(&smx[2], mx2);
    __syncthreads();
    if (tid < 3) {
        atomicMin(&minc[b * 3 + tid], smn[tid]);
        atomicMax(&maxc[b * 3 + tid], smx[tid]);
    }
}

__global__ __launch_bounds__(32) void cmax_kernel(const int* __restrict__ minc,
                                                  const int* __restrict__ maxc,
                                                  long long* __restrict__ params) {
    if (threadIdx.x != 0) return;
#pragma unroll
    for (int a = 0; a < 3; ++a) {
        int m = INT_MIN;
#pragma unroll
        for (int b = 0; b < B_CLOUDS; ++b)
            m = max(m, maxc[b * 3 + a] - minc[b * 3 + a]);
        params[a] = (long long)m + 1;                 // cmax[a+1]
    }
}

// ---------------------------------------------------------------- key generation
__global__ __launch_bounds__(256) void key_kernel(const float4* __restrict__ pts4,
                                                  const int* __restrict__ minc,
                                                  const long long* __restrict__ params,
                                                  u64* __restrict__ keys) {
    unsigned i = blockIdx.x * 256u + threadIdx.x;
    if (i >= (unsigned)PADDED_N) return;
    if (i >= (unsigned)BN) { keys[i] = ~0ull; return; }   // pad keys sort last
    int b = (int)(i / (unsigned)N_PTS);
    float4 p = pts4[i];
    long long c1 = params[0], c2 = params[1], c3 = params[2];
    long long cx = (int)(p.x / VSIZE) - minc[b * 3 + 0];
    long long cy = (int)(p.y / VSIZE) - minc[b * 3 + 1];
    long long cz = (int)(p.z / VSIZE) - minc[b * 3 + 2];
    long long vkey = (((long long)b * c1 + cx) * c2 + cy) * c3 + cz;
    keys[i] = ((u64)vkey << IDX_BITS) | (u64)i;
}

// ---------------------------------------------------------------- radix: histogram
__global__ __launch_bounds__(256) void hist_kernel(const u64* __restrict__ keys,
                                                   u32* __restrict__ hist,
                                                   int shift) {
    __shared__ u32 lh[RADIX];
    const int tid = threadIdx.x;
#pragma unroll
    for (int k = 0; k < RADIX / 256; ++k) lh[tid + k * 256] = 0;
    __syncthreads();
    size_t base = (size_t)blockIdx.x * TILE;
#pragma unroll
    for (int it = 0; it < TILE / 256; ++it) {
        size_t i = base + (size_t)it * 256 + tid;
        if (blockIdx.x + 1 < NUM_TILES)
            __builtin_prefetch(&keys[i + TILE], 0, 0);   // global_prefetch_b8
        u32 d = (u32)(keys[i] >> shift) & (RADIX - 1);
        atomicAdd(&lh[d], 1u);
    }
    __syncthreads();
#pragma unroll
    for (int k = 0; k < RADIX / 256; ++k) {
        int d = tid + k * 256;
        hist[(size_t)d * NUM_TILES + blockIdx.x] = lh[d];
    }
}

// ---------------------------------------------------------------- radix: row scan
__global__ __launch_bounds__(256) void rowscan_kernel(u32* __restrict__ hist,
                                                      u32* __restrict__ rowTotal) {
    __shared__ u32 s[256];
    const int tid = threadIdx.x;
    u32* row = hist + (size_t)blockIdx.x * NUM_TILES;
    u32 carry = 0;
    for (int ch = 0; ch < 8; ++ch) {               // 8*256 = 2048 >= 1954
        int idx = ch * 256 + tid;
        u32 v = (idx < NUM_TILES) ? row[idx] : 0u;
        s[tid] = v;
        __syncthreads();
        for (int off = 1; off < 256; off <<= 1) {
            u32 t = (tid >= off) ? s[tid - off] : 0u;
            __syncthreads();
            s[tid] += t;
            __syncthreads();
        }
        u32 incl = s[tid];
        u32 total = s[255];
        if (idx < NUM_TILES) row[idx] = incl - v + carry;
        carry += total;
        __syncthreads();
    }
    if (tid == 0) rowTotal[blockIdx.x] = carry;
}

// ---------------------------------------------------------------- radix: digit scan
__global__ __launch_bounds__(256) void digitscan_kernel(const u32* __restrict__ rowTotal,
                                                        u32* __restrict__ digitBase) {
    __shared__ u32 s[256];
    const int tid = threadIdx.x;
    u32 acc = 0;
#pragma unroll
    for (int k = 0; k < RADIX / 256; ++k) acc += rowTotal[tid * (RADIX / 256) + k];
    s[tid] = acc;
    __syncthreads();
    for (int off = 1; off < 256; off <<= 1) {
        u32 t = (tid >= off) ? s[tid - off] : 0u;
        __syncthreads();
        s[tid] += t;
        __syncthreads();
    }
    u32 run = s[tid] - acc;
#pragma unroll
    for (int k = 0; k < RADIX / 256; ++k) {
        int d = tid * (RADIX / 256) + k;
        u32 v = rowTotal[d];
        digitBase[d] = run;
        run += v;
    }
}

// ---------------------------------------------------------------- radix: scatter
__global__ __launch_bounds__(SC_T) void scatter_kernel(const u64* __restrict__ keyIn,
                                                       u64* __restrict__ keyOut,
                                                       const u32* __restrict__ hist,
                                                       const u32* __restrict__ digitBase,
                                                       int shift) {
    __shared__ u64 tk[TILE];            // 32 KB: tile keys staged via TDM
    __shared__ u32 cnt[RADIX * SC_W];   // 64 KB: per-(digit, wave) counters
    __shared__ u32 sbase[RADIX];        // 16 KB: global base per digit for tile
    __shared__ u32 sp[SC_T];

    const int tid  = threadIdx.x;
    const int wv   = tid >> 5;
    const int lane = tid & 31;
    const size_t tileBase = (size_t)blockIdx.x * TILE;

#ifdef HAVE_TDM
    if (tid < 32) {
        // Tensor Data Mover: DMA TILE u64 keys -> LDS (D# per ISA 8.3/8.4).
        u64 ga  = (u64)(uintptr_t)(keyIn + tileBase);
        u32 lds = (u32)(uintptr_t)(&tk[0]);
        u32x4 g0 = { 1u,                                   // count = 1
                     lds,                                  // lds_addr
                     (u32)(ga & 0xFFFFFFFFull),            // global_addr[31:0]
                     (u32)((ga >> 32) & 0x01FFFFFFull) | (2u << 30) }; // [56:32] | type=2
        i32x8 g1 = { (int)(3u << 16),          // data_size = 3 (8 bytes)
                     (int)((u32)TILE << 16),   // tensor_dim0 lo16
                     (int)(1u << 16),          // tensor_dim0 hi16 | tensor_dim1 lo16 = 1
                     (int)((u32)TILE << 16),   // tensor_dim1 hi16 | tile_dim0 = TILE
                     1,                        // tile_dim1 = 1, tile_dim2 = 0
                     TILE,                     // tensor_dim0_stride lo32
                     (int)((u32)TILE << 16),   // stride hi16 | tensor_dim1_stride lo16
                     0 };
        i32x4 g2 = { 0, 0, 0, 0 };
        i32x4 g3 = { 0, 0, 0, 0 };
        i32x8 g4 = { 0, 0, 0, 0, 0, 0, 0, 0 };   // clang-23 lane: 6-arg form
        __builtin_amdgcn_tensor_load_to_lds(g0, g1, g2, g3, g4, 0);
        __builtin_amdgcn_s_wait_tensorcnt(0);
    }
#else
#pragma unroll
    for (int k = 0; k < TILE / SC_T; ++k) {
        int l = tid + k * SC_T;
        tk[l] = keyIn[tileBase + l];
    }
#endif

    // zero counters + stage per-digit global bases for this tile
#pragma unroll
    for (int k = 0; k < (RADIX * SC_W) / SC_T; ++k) cnt[tid + k * SC_T] = 0u;
#pragma unroll
    for (int k = 0; k < RADIX / SC_T; ++k) {
        int d = tid + k * SC_T;
        sbase[d] = digitBase[d] + hist[(size_t)d * NUM_TILES + blockIdx.x];
    }
    __syncthreads();

    // phase 1: per-(digit,wave) counts (order irrelevant here)
    const int waveBase = wv * (32 * SC_I);
#pragma unroll
    for (int it = 0; it < SC_I; ++it) {
        int l = waveBase + it * 32 + lane;
        u32 d = (u32)(tk[l] >> shift) & (RADIX - 1);
        atomicAdd(&cnt[d * SC_W + wv], 1u);
    }
    __syncthreads();

    // phase 2: exclusive scan of cnt[] (digit-major, wave-minor) in place
    {
        const int chunk = (RADIX * SC_W) / SC_T;      // 128
        u32 acc = 0;
        int base = tid * chunk;
#pragma unroll 4
        for (int k = 0; k < chunk; ++k) acc += cnt[base + k];
        sp[tid] = acc;
        __syncthreads();
        for (int off = 1; off < SC_T; off <<= 1) {
            u32 t = (tid >= off) ? sp[tid - off] : 0u;
            __syncthreads();
            sp[tid] += t;
            __syncthreads();
        }
        u32 run = sp[tid] - acc;
#pragma unroll 4
        for (int k = 0; k < chunk; ++k) {
            u32 v = cnt[base + k];
            cnt[base + k] = run;
            run += v;
        }
    }
    __syncthreads();

    // phase 3: deterministic (stable) ranked scatter, wave32 ballot matching
    const u32 ltmask = (1u << lane) - 1u;
#pragma unroll 4
    for (int it = 0; it < SC_I; ++it) {
        int l = waveBase + it * 32 + lane;
        u64 key = tk[l];
        u32 d = (u32)(key >> shift) & (RADIX - 1);
        u32 peers = 0xFFFFFFFFu;
#pragma unroll
        for (int b = 0; b < RADIX_BITS; ++b) {
            u32 bit = (d >> b) & 1u;
            u32 bal = (u32)__ballot(bit);
            peers &= bit ? bal : ~bal;
        }
        int leader = __builtin_ctz(peers);
        u32 old = 0;
        if (lane == leader)
            old = atomicAdd(&cnt[d * SC_W + wv], (u32)__popc(peers));
        old = (u32)__shfl((int)old, leader, 32);
        u32 rank = old + (u32)__popc(peers & ltmask);
        keyOut[sbase[d] + rank] = key;
    }
}

// ---------------------------------------------------------------- output: pad fill
__global__ __launch_bounds__(256) void fill_kernel(const u64* __restrict__ sorted,
                                                   const float4* __restrict__ pts4,
                                                   const int* __restrict__ minc,
                                                   float4* __restrict__ outPts,
                                                   long long* __restrict__ outCoords) {
    unsigned j = blockIdx.x * 256u + threadIdx.x;
    if (j >= (unsigned)BN) return;
    u64 k0 = sorted[0];                         // idx_sort[0]: reference pad pick
    u32 idx = (u32)(k0 & IDX_MASK);
    int b = (int)(idx / (u32)N_PTS);
    float4 p = pts4[idx];
    long long cx = (int)(p.x / VSIZE) - minc[b * 3 + 0];
    long long cy = (int)(p.y / VSIZE) - minc[b * 3 + 1];
    long long cz = (int)(p.z / VSIZE) - minc[b * 3 + 2];
    outPts[j] = p;
    long long* oc = outCoords + (size_t)j * 4;
    oc[0] = b; oc[1] = cx; oc[2] = cy; oc[3] = cz;
}

// ---------------------------------------------------------------- compaction
__global__ __launch_bounds__(256) void bcount_kernel(const u64* __restrict__ sorted,
                                                     u32* __restrict__ bcount) {
    __shared__ u32 c;
    const int tid = threadIdx.x;
    if (tid == 0) c = 0;
    __syncthreads();
    size_t base = (size_t)blockIdx.x * TILE + (size_t)tid * 16;
    u32 local = 0;
#pragma unroll
    for (int k = 0; k < 16; ++k) {
        size_t i = base + k;
        if (i < (size_t)BN) {
            u64 v = sorted[i] >> IDX_BITS;
            bool f = (i == 0) || ((sorted[i - 1] >> IDX_BITS) != v);
            local += f ? 1u : 0u;
        }
    }
    atomicAdd(&c, local);
    __syncthreads();
    if (tid == 0) bcount[blockIdx.x] = c;
}

__global__ __launch_bounds__(256) void bscan_kernel(const u32* __restrict__ bcount,
                                                    u32* __restrict__ bbase,
                                                    long long* __restrict__ outNum) {
    __shared__ u32 s[256];
    const int tid = threadIdx.x;
    u32 carry = 0;
    for (int ch = 0; ch < 8; ++ch) {
        int idx = ch * 256 + tid;
        u32 v = (idx < NUM_TILES) ? bcount[idx] : 0u;
        s[tid] = v;
        __syncthreads();
        for (int off = 1; off < 256; off <<= 1) {
            u32 t = (tid >= off) ? s[tid - off] : 0u;
            __syncthreads();
            s[tid] += t;
            __syncthreads();
        }
        u32 incl = s[tid];
        u32 total = s[255];
        if (idx < NUM_TILES) bbase[idx] = incl - v + carry;
        carry += total;
        __syncthreads();
    }
    if (tid == 0) *outNum = (long long)carry;   // num_unique (int64)
}

__global__ __launch_bounds__(256) void compact_kernel(const u64* __restrict__ sorted,
                                                      const float4* __restrict__ pts4,
                                                      const int* __restrict__ minc,
                                                      const u32* __restrict__ bbase,
                                                      float4* __restrict__ outPts,
                                                      long long* __restrict__ outCoords) {
    __shared__ u32 sp[256];
    const int tid = threadIdx.x;
    size_t base = (size_t)blockIdx.x * TILE + (size_t)tid * 16;

    u32 myc = 0;
#pragma unroll
    for (int k = 0; k < 16; ++k) {
        size_t i = base + k;
        if (i < (size_t)BN) {
            u64 v = sorted[i] >> IDX_BITS;
            myc += ((i == 0) || ((sorted[i - 1] >> IDX_BITS) != v)) ? 1u : 0u;
        }
    }
    sp[tid] = myc;
    __syncthreads();
    for (int off = 1; off < 256; off <<= 1) {
        u32 t = (tid >= off) ? sp[tid - off] : 0u;
        __syncthreads();
        sp[tid] += t;
        __syncthreads();
    }
    u32 run = bbase[blockIdx.x] + sp[tid] - myc;

#pragma unroll
    for (int k = 0; k < 16; ++k) {
        size_t i = base + k;
        if (i < (size_t)BN) {
            u64 key = sorted[i];
            u64 v = key >> IDX_BITS;
            if ((i == 0) || ((sorted[i - 1] >> IDX_BITS) != v)) {
                u32 idx = (u32)(key & IDX_MASK);
                int b = (int)(idx / (u32)N_PTS);
                float4 p = pts4[idx];
                long long cx = (int)(p.x / VSIZE) - minc[b * 3 + 0];
                long long cy = (int)(p.y / VSIZE) - minc[b * 3 + 1];
                long long cz = (int)(p.z / VSIZE) - minc[b * 3 + 2];
                outPts[run] = p;
                long long* oc = outCoords + (size_t)run * 4;
                oc[0] = b; oc[1] = cx; oc[2] = cy; oc[3] = cz;
                ++run;
            }
        }
    }
}

// ---------------------------------------------------------------- launcher
extern "C" void kernel_launch(void* const* d_in, const int* in_sizes, int n_in,
                              void* d_out, int out_size, void* d_ws, size_t ws_size,
                              hipStream_t stream) {
    (void)in_sizes; (void)n_in; (void)out_size; (void)ws_size;
    const float4* pts4 = (const float4*)d_in[0];

    // workspace carve-up (16B aligned sections)
    char* ws = (char*)d_ws;
    size_t o = 0;
    auto carve = [&](size_t bytes) -> void* {
        void* p = ws + o;
        o += (bytes + 15) & ~(size_t)15;
        return p;
    };
    u64* keysA      = (u64*)carve((size_t)PADDED_N * 8);
    u64* keysB      = (u64*)carve((size_t)PADDED_N * 8);
    u32* hist       = (u32*)carve((size_t)RADIX * NUM_TILES * 4);
    u32* rowTotal   = (u32*)carve((size_t)RADIX * 4);
    u32* digitBase  = (u32*)carve((size_t)RADIX * 4);
    int* minc       = (int*)carve((size_t)B_CLOUDS * 3 * 4);
    int* maxc       = (int*)carve((size_t)B_CLOUDS * 3 * 4);
    long long* prm  = (long long*)carve(4 * 8);
    u32* bcount     = (u32*)carve((size_t)NUM_TILES * 4);
    u32* bbase      = (u32*)carve((size_t)NUM_TILES * 4);

    // output regions: [points f32 BN*4][coords i64 BN*4][num_unique i64]
    float4*    outPts    = (float4*)d_out;
    long long* outCoords = (long long*)((float*)d_out + (size_t)BN * 4);
    long long* outNum    = outCoords + (size_t)BN * 4;

    // 1) per-cloud coordinate min/max
    init_kernel<<<1, 64, 0, stream>>>(minc, maxc);
    dim3 mmGrid((N_PTS + TILE - 1) / TILE, B_CLOUDS);
    minmax_kernel<<<mmGrid, 256, 0, stream>>>(pts4, minc, maxc);
    cmax_kernel<<<1, 32, 0, stream>>>(minc, maxc, prm);

    // 2) composite keys (+ pad to tile multiple with 0xFF.. keys)
    key_kernel<<<(PADDED_N + 255) / 256, 256, 0, stream>>>(pts4, minc, prm, keysA);

    // 3) 5-pass LSD radix sort on low 60 bits (56 meaningful)
    u64* src = keysA;
    u64* dst = keysB;
    for (int p = 0; p < 5; ++p) {
        int shift = p * RADIX_BITS;
        hist_kernel<<<NUM_TILES, 256, 0, stream>>>(src, hist, shift);
        rowscan_kernel<<<RADIX, 256, 0, stream>>>(hist, rowTotal);
        digitscan_kernel<<<1, 256, 0, stream>>>(rowTotal, digitBase);
        scatter_kernel<<<NUM_TILES, SC_T, 0, stream>>>(src, dst, hist, digitBase, shift);
        u64* t = src; src = dst; dst = t;
    }
    u64* sorted = src;   // after 5 swaps: keysB

    // 4) outputs: pad-fill then ranked compaction + num_unique
    fill_kernel<<<(BN + 255) / 256, 256, 0, stream>>>(sorted, pts4, minc, outPts, outCoords);
    bcount_kernel<<<NUM_TILES, 256, 0, stream>>>(sorted, bcount);
    bscan_kernel<<<1, 256, 0, stream>>>(bcount, bbase, outNum);
    compact_kernel<<<NUM_TILES, 256, 0, stream>>>(sorted, pts4, minc, bbase, outPts, outCoords);
}